// GATModel_48112223649848
// MI455X (gfx1250) — compile-verified
//
#include <hip/hip_runtime.h>

// ---------------------------------------------------------------------------
// GATv2 (2 layers) for MI455X / gfx1250.
// Memory-bound (~300-400MB traffic @ 23.3 TB/s; all per-node state is
// L2-resident in the 192MB L2). Projections use V_WMMA_F32_16X16X4_F32
// (exact fp32 matrix math, wave32 layout).
// ---------------------------------------------------------------------------

typedef __attribute__((ext_vector_type(2))) float v2f;
typedef __attribute__((ext_vector_type(8))) float v8f;

#define NNODES 100000
#define NEDGES 3200000
#define ETOT   (NEDGES + NNODES)   // edges + self loops
#define FIN    256
#define NEG_SLOPE 0.2f

// Monotone float <-> ordered-int mapping: integer atomicMax == float max.
__device__ __forceinline__ int fkey(float f) {
    int i = __float_as_int(f);
    return i >= 0 ? i : (i ^ 0x7fffffff);
}
__device__ __forceinline__ float keyf(int k) {
    return __int_as_float(k >= 0 ? k : (k ^ 0x7fffffff));
}

// ---------------------------------------------------------------------------
// Layer-1 projection: z1[N,16] = x[N,256] @ [Wl1 | Wr1] + [bl1 | br1]
// One wave computes a 16x16 tile via 64 x V_WMMA_F32_16X16X4_F32.
// A (16x4 f32): lane l holds M=l%16, K=(l/16)*2 + {0,1}
// B (4x16 f32): lane l holds N=l%16, K=(l/16)*2 + {0,1}
// C/D (16x16):  lane l, VGPR v holds M=v+(l/16)*8, N=l%16
// ---------------------------------------------------------------------------
__global__ __launch_bounds__(256) void gat_gemm1(
    const float* __restrict__ x,
    const float* __restrict__ Wl, const float* __restrict__ bl,
    const float* __restrict__ Wr, const float* __restrict__ br,
    float* __restrict__ z)
{
    __shared__ float Ws[FIN * 16];          // fused [256][16] weights, 16 KB LDS
    const int tid = threadIdx.x;
    for (int i = tid; i < FIN * 16; i += 256) {
        int k = i >> 4, n = i & 15;
        Ws[i] = (n < 8) ? Wl[k * 8 + n] : Wr[k * 8 + (n - 8)];
    }
    __syncthreads();

    const int lane = tid & 31;
    const int wave = tid >> 5;
    const int hi   = lane >> 4;             // half-wave selector
    const int lo   = lane & 15;
    const int row0 = (blockIdx.x * 8 + wave) * 16;

    int rowA = row0 + lo;                   // A-operand row for this lane
    if (rowA >= NNODES) rowA = NNODES - 1;  // clamp: keep EXEC all-1s for WMMA
    const float* __restrict__ xrow = x + (long long)rowA * FIN;

    v8f c = {};
    #pragma unroll 8
    for (int k = 0; k < FIN; k += 4) {
        const int kb = k + hi * 2;
        v2f a, b;
        a[0] = xrow[kb];
        a[1] = xrow[kb + 1];
        b[0] = Ws[kb * 16 + lo];
        b[1] = Ws[(kb + 1) * 16 + lo];
        c = __builtin_amdgcn_wmma_f32_16x16x4_f32(
                false, a, false, b, (short)0, c, false, false);
    }

    const float bias = (lo < 8) ? bl[lo] : br[lo - 8];
    float* __restrict__ zc = z + (long long)(row0 + hi * 8) * 16 + lo;
    if (row0 + 16 <= NNODES) {              // full tile: unguarded stores
        #pragma unroll
        for (int v = 0; v < 8; v++) zc[v * 16] = c[v] + bias;
    } else {                                // last partial tile only
        #pragma unroll
        for (int v = 0; v < 8; v++) {
            if (row0 + hi * 8 + v < NNODES) zc[v * 16] = c[v] + bias;
        }
    }
}

// ---------------------------------------------------------------------------
// Layer-2 projection: z2[N,32] = h[N,8] @ [Wl2 | Wr2] + [bl2 | br2]
// K=8 -> two WMMA k-steps; two 16-wide N tiles (lin_l cols, lin_r cols).
// ---------------------------------------------------------------------------
__global__ __launch_bounds__(256) void gat_gemm2(
    const float* __restrict__ h,
    const float* __restrict__ Wl, const float* __restrict__ bl,
    const float* __restrict__ Wr, const float* __restrict__ br,
    float* __restrict__ z)
{
    __shared__ float Ws[2 * 8 * 16];        // [tile][k][n], 1 KB LDS
    const int tid = threadIdx.x;
    if (tid < 128) Ws[tid] = Wl[tid];
    else           Ws[tid] = Wr[tid - 128];
    __syncthreads();

    const int lane = tid & 31;
    const int wave = tid >> 5;
    const int hi   = lane >> 4;
    const int lo   = lane & 15;
    const int row0 = (blockIdx.x * 8 + wave) * 16;

    int rowA = row0 + lo;
    if (rowA >= NNODES) rowA = NNODES - 1;
    const float* __restrict__ hrow = h + (long long)rowA * 8;

    v8f c0 = {}, c1 = {};
    #pragma unroll
    for (int k = 0; k < 8; k += 4) {
        const int kb = k + hi * 2;
        v2f a, b0, b1;
        a[0] = hrow[kb];
        a[1] = hrow[kb + 1];
        b0[0] = Ws[kb * 16 + lo];
        b0[1] = Ws[(kb + 1) * 16 + lo];
        b1[0] = Ws[128 + kb * 16 + lo];
        b1[1] = Ws[128 + (kb + 1) * 16 + lo];
        c0 = __builtin_amdgcn_wmma_f32_16x16x4_f32(false, a, false, b0, (short)0, c0, false, false);
        c1 = __builtin_amdgcn_wmma_f32_16x16x4_f32(false, a, false, b1, (short)0, c1, false, false);
    }

    const float biasL = bl[lo];
    const float biasR = br[lo];
    float* __restrict__ zc = z + (long long)(row0 + hi * 8) * 32 + lo;
    if (row0 + 16 <= NNODES) {
        #pragma unroll
        for (int v = 0; v < 8; v++) {
            zc[v * 32]      = c0[v] + biasL;
            zc[v * 32 + 16] = c1[v] + biasR;
        }
    } else {
        #pragma unroll
        for (int v = 0; v < 8; v++) {
            if (row0 + hi * 8 + v < NNODES) {
                zc[v * 32]      = c0[v] + biasL;
                zc[v * 32 + 16] = c1[v] + biasR;
            }
        }
    }
}

// ---------------------------------------------------------------------------
// Edge helpers: b128 gathers of the [xl(D) | xr(D)] rows (L2-resident).
// ---------------------------------------------------------------------------
__device__ __forceinline__ void edge_pair(const int* __restrict__ ei, int e,
                                          int& s, int& d) {
    if (e < NEDGES) { s = ei[e]; d = ei[NEDGES + e]; }
    else            { s = d = e - NEDGES; }            // self loop
}

template <int D, int STR>
__device__ __forceinline__ float edge_logit(const float* __restrict__ z,
                                            int s, int d,
                                            const float* __restrict__ att) {
    const float4* __restrict__ xl = (const float4*)(z + (long long)s * STR);
    const float4* __restrict__ xr = (const float4*)(z + (long long)d * STR + D);
    float acc = 0.f;
    #pragma unroll
    for (int q = 0; q < D / 4; q++) {
        const float4 l = xl[q];
        const float4 r = xr[q];
        float vv[4] = { l.x + r.x, l.y + r.y, l.z + r.z, l.w + r.w };
        #pragma unroll
        for (int t = 0; t < 4; t++) {
            float v = vv[t];
            v = v > 0.f ? v : NEG_SLOPE * v;
            acc += v * att[q * 4 + t];
        }
    }
    return acc;
}

// Pass 1: logits + per-destination segment max (ordered-int atomicMax).
template <int D, int STR>
__global__ __launch_bounds__(256) void gat_edge_max(
    const int* __restrict__ ei, const float* __restrict__ z,
    const float* __restrict__ att,
    float* __restrict__ sbuf, int* __restrict__ mkey)
{
    const int e = blockIdx.x * 256 + threadIdx.x;
    if (e >= ETOT) return;
    int s, d;
    edge_pair(ei, e, s, d);
    const float acc = edge_logit<D, STR>(z, s, d, att);
    sbuf[e] = acc;
    atomicMax(&mkey[d], fkey(acc));
}

// Pass 2: ex = exp(s - max[dst]); segment sum.
__global__ __launch_bounds__(256) void gat_edge_exp(
    const int* __restrict__ ei, float* __restrict__ sbuf,
    const int* __restrict__ mkey, float* __restrict__ denom)
{
    const int e = blockIdx.x * 256 + threadIdx.x;
    if (e >= ETOT) return;
    const int d = (e < NEDGES) ? ei[NEDGES + e] : (e - NEDGES);
    const float ex = __expf(sbuf[e] - keyf(mkey[d]));
    sbuf[e] = ex;
    atomicAdd(&denom[d], ex);
}

// Pass 3: agg[dst] += (ex / denom[dst]) * xl[src]
template <int D, int STR>
__global__ __launch_bounds__(256) void gat_edge_agg(
    const int* __restrict__ ei, const float* __restrict__ z,
    const float* __restrict__ sbuf, const float* __restrict__ denom,
    float* __restrict__ agg)
{
    const int e = blockIdx.x * 256 + threadIdx.x;
    if (e >= ETOT) return;
    int s, d;
    edge_pair(ei, e, s, d);
    const float alpha = sbuf[e] / denom[d];
    const float4* __restrict__ xl = (const float4*)(z + (long long)s * STR);
    float* __restrict__ ap = agg + (long long)d * D;
    #pragma unroll
    for (int q = 0; q < D / 4; q++) {
        const float4 l = xl[q];
        atomicAdd(&ap[q * 4 + 0], alpha * l.x);
        atomicAdd(&ap[q * 4 + 1], alpha * l.y);
        atomicAdd(&ap[q * 4 + 2], alpha * l.z);
        atomicAdd(&ap[q * 4 + 3], alpha * l.w);
    }
}

// ---------------------------------------------------------------------------
// Fused per-layer init: mkey = -inf key, denom = 0, agg = 0.
// ---------------------------------------------------------------------------
__global__ __launch_bounds__(256) void gat_init(
    int* __restrict__ mkey, float* __restrict__ denom,
    float* __restrict__ agg, int nAgg)
{
    const int i = blockIdx.x * 256 + threadIdx.x;
    if (i < NNODES) { mkey[i] = (int)0x80000000; denom[i] = 0.f; }
    if (i < nAgg)   agg[i] = 0.f;
}

// mode relu=1: out = relu(in + bias[j]) ; relu=0: out = in + bias[j]
__global__ __launch_bounds__(256) void gat_bias_act(
    const float* __restrict__ in, const float* __restrict__ bias,
    float* __restrict__ out, int n, int d, int relu)
{
    const int i = blockIdx.x * 256 + threadIdx.x;
    if (i >= n) return;
    float v = in[i] + bias[i % d];
    if (relu) v = fmaxf(v, 0.f);
    out[i] = v;
}

// ---------------------------------------------------------------------------
// Host-side launcher
// ---------------------------------------------------------------------------
extern "C" void kernel_launch(void* const* d_in, const int* in_sizes, int n_in,
                              void* d_out, int out_size, void* d_ws, size_t ws_size,
                              hipStream_t stream)
{
    const float* x     = (const float*)d_in[0];
    const int*   ei    = (const int*)  d_in[1];
    const float* Wl1   = (const float*)d_in[2];
    const float* bl1   = (const float*)d_in[3];
    const float* Wr1   = (const float*)d_in[4];
    const float* br1   = (const float*)d_in[5];
    const float* att1  = (const float*)d_in[6];
    const float* bias1 = (const float*)d_in[7];
    const float* Wl2   = (const float*)d_in[8];
    const float* bl2   = (const float*)d_in[9];
    const float* Wr2   = (const float*)d_in[10];
    const float* br2   = (const float*)d_in[11];
    const float* att2  = (const float*)d_in[12];
    const float* bias2 = (const float*)d_in[13];
    float* out = (float*)d_out;

    // workspace carve-up (floats)
    float* ws    = (float*)d_ws;
    float* z1    = ws;                          // N*16  [xl1|xr1]
    float* z2    = z1   + (size_t)NNODES * 16;  // N*32  [xl2|xr2]
    float* h     = z2   + (size_t)NNODES * 32;  // N*8
    float* agg   = h    + (size_t)NNODES * 8;   // N*16 (layer1 uses N*8)
    float* sbuf  = agg  + (size_t)NNODES * 16;  // ETOT  per-edge logits/exps
    int*   mkey  = (int*)  (sbuf + (size_t)ETOT);  // N
    float* denom = (float*)(mkey + NNODES);        // N

    const dim3 blk(256);
    const int gemmGrid = (NNODES + 127) / 128;      // 8 waves x 16 rows / block
    const int edgeGrid = (ETOT + 255) / 256;
    const int nGrid8   = (NNODES * 8  + 255) / 256;
    const int nGrid16  = (NNODES * 16 + 255) / 256;

    // ---------------- layer 1 ----------------
    gat_gemm1<<<gemmGrid, blk, 0, stream>>>(x, Wl1, bl1, Wr1, br1, z1);
    gat_init<<<nGrid8, blk, 0, stream>>>(mkey, denom, agg, NNODES * 8);
    gat_edge_max<8, 16><<<edgeGrid, blk, 0, stream>>>(ei, z1, att1, sbuf, mkey);
    gat_edge_exp<<<edgeGrid, blk, 0, stream>>>(ei, sbuf, mkey, denom);
    gat_edge_agg<8, 16><<<edgeGrid, blk, 0, stream>>>(ei, z1, sbuf, denom, agg);
    gat_bias_act<<<nGrid8, blk, 0, stream>>>(agg, bias1, h, NNODES * 8, 8, 1);

    // ---------------- layer 2 ----------------
    gat_gemm2<<<gemmGrid, blk, 0, stream>>>(h, Wl2, bl2, Wr2, br2, z2);
    gat_init<<<nGrid16, blk, 0, stream>>>(mkey, denom, agg, NNODES * 16);
    gat_edge_max<16, 32><<<edgeGrid, blk, 0, stream>>>(ei, z2, att2, sbuf, mkey);
    gat_edge_exp<<<edgeGrid, blk, 0, stream>>>(ei, sbuf, mkey, denom);
    gat_edge_agg<16, 32><<<edgeGrid, blk, 0, stream>>>(ei, z2, sbuf, denom, agg);
    gat_bias_act<<<nGrid16, blk, 0, stream>>>(agg, bias2, out, NNODES * 16, 16, 0);
}